// QuadraticFieldEnergyReadout_13872744366319
// MI455X (gfx1250) — compile-verified
//
#include <hip/hip_runtime.h>
#include <hip/hip_bf16.h>

typedef __attribute__((ext_vector_type(2))) float v2f;
typedef __attribute__((ext_vector_type(8))) float v8f;

#define IS3 0.57735026918962576451f   // 1/sqrt(3)
#define FINAL_SCALE (1.0f / 8192.0f)  // A_TP * A_DN = 1/sqrt(2048*32768)

// ---------------------------------------------------------------------------
// Kernel 1: contract the four (32,32,128) tensors with coefficient outer
// products -> P (16x128, rows 0..4 used) and Q (16x128, rows 0..3 used).
// ws layout (floats): P @ 0, Q @ 2048, R @ 4096, G @ 6144.
// ---------------------------------------------------------------------------
__global__ void precompute_PQ(const float* __restrict__ Wss, const float* __restrict__ Wvv,
                              const float* __restrict__ Wsv, const float* __restrict__ Wvs,
                              const float* __restrict__ wqa_s, const float* __restrict__ bqa,
                              const float* __restrict__ wqa_v, const float* __restrict__ wva_s,
                              const float* __restrict__ bva, const float* __restrict__ wva_v,
                              float* __restrict__ ws) {
  const int w   = threadIdx.x;   // 0..127 (output channel)
  const int blk = blockIdx.x;    // 0..3 (which W tensor)
  __shared__ float sA0[32], sA1[32], sB0[32], sB1[32];

  const float *W, *a0v, *a1v, *b0v, *b1v;
  if (blk == 0)      { W = Wss; a0v = wqa_s; a1v = bqa;     b0v = wva_s; b1v = bva;     }
  else if (blk == 1) { W = Wvv; a0v = wqa_v; a1v = nullptr; b0v = wva_v; b1v = nullptr; }
  else if (blk == 2) { W = Wsv; a0v = wqa_s; a1v = bqa;     b0v = wva_v; b1v = nullptr; }
  else               { W = Wvs; a0v = wqa_v; a1v = nullptr; b0v = wva_s; b1v = bva;     }

  if (w < 32) {
    sA0[w] = a0v[w];
    sA1[w] = a1v ? a1v[w] : 0.0f;
    sB0[w] = b0v[w];
    sB1[w] = b1v ? b1v[w] : 0.0f;
  }
  __syncthreads();

  float acc00 = 0.f, acc01 = 0.f, acc10 = 0.f, acc11 = 0.f;
  for (int u = 0; u < 32; ++u) {
    const float a0 = sA0[u], a1 = sA1[u];
    const float* Wu = W + u * 32 * 128 + w;
#pragma unroll 4
    for (int v = 0; v < 32; ++v) {
      const float wv = Wu[v * 128];            // coalesced across the 128 lanes
      const float b0 = sB0[v], b1 = sB1[v];
      acc00 = fmaf(a0 * b0, wv, acc00);
      acc01 = fmaf(a0 * b1, wv, acc01);
      acc10 = fmaf(a1 * b0, wv, acc10);
      acc11 = fmaf(a1 * b1, wv, acc11);
    }
  }

  float* P = ws;
  float* Q = ws + 2048;
  if (blk == 0) {          // P1=(wqa_s,wva_s) P2=(wqa_s,bva) P3=(bqa,wva_s) P4=(bqa,bva)
    P[0*128 + w] = acc00;  P[1*128 + w] = acc01;
    P[2*128 + w] = acc10;  P[3*128 + w] = acc11;
  } else if (blk == 1) {   // P5=(wqa_v,wva_v)
    P[4*128 + w] = acc00;
    for (int r = 5; r < 16; ++r) P[r*128 + w] = 0.f;   // pad rows for the WMMA gemm
  } else if (blk == 2) {   // Q1=(wqa_s,wva_v) Q2=(bqa,wva_v)
    Q[0*128 + w] = acc00;  Q[1*128 + w] = acc10;
  } else {                 // Q3=(wqa_v,wva_s) Q4=(wqa_v,bva)
    Q[2*128 + w] = acc00;  Q[3*128 + w] = acc01;
    for (int r = 4; r < 16; ++r) Q[r*128 + w] = 0.f;
  }
}

// ---------------------------------------------------------------------------
// Kernel 2: R(16x128) = P @ Wd_ss,  G(16x128) = Q @ Wd_vv, via
// v_wmma_f32_16x16x4_f32. One wave per 16x16 output tile (8 N-tiles x 2 gemms
// = 16 waves = 512 threads). fp32 A/B layouts per ISA 7.12.2:
//   A 16x4 : lanes 0-15 -> rows, vgpr0/1 = K0/K1; lanes 16-31 = K2/K3
//   B 4x16 : lanes 0-15 -> cols, vgpr0/1 = K0/K1; lanes 16-31 = K2/K3
//   D 16x16: vgpr r = row r (lanes 0-15) / row r+8 (lanes 16-31)
// ---------------------------------------------------------------------------
__global__ void small_gemm_wmma(const float* __restrict__ ws_in,
                                const float* __restrict__ Wd_ss,
                                const float* __restrict__ Wd_vv,
                                float* __restrict__ ws_out) {
  const int tid  = threadIdx.x;
  const int wave = tid >> 5;
  const int lane = tid & 31;
  const int half = lane >> 4;     // 0: K pair {k,k+1}; 1: {k+2,k+3}
  const int lr   = lane & 15;
  const int sel  = wave >> 3;     // 0: R = P @ Wd_ss; 1: G = Q @ Wd_vv
  const int col0 = (wave & 7) * 16;

  const float* A = ws_in + sel * 2048;     // 16x128 row-major
  const float* B = sel ? Wd_vv : Wd_ss;    // 128x128 row-major
  float*       D = ws_out + sel * 2048;    // 16x128 row-major

  v8f c = {};
  for (int k = 0; k < 128; k += 4) {
    const int ka = k + 2 * half;
    v2f a, b;
    a.x = A[lr * 128 + ka];
    a.y = A[lr * 128 + ka + 1];
    b.x = B[ka * 128 + col0 + lr];
    b.y = B[(ka + 1) * 128 + col0 + lr];
    c = __builtin_amdgcn_wmma_f32_16x16x4_f32(false, a, false, b, (short)0, c, false, false);
  }
#pragma unroll
  for (int r = 0; r < 8; ++r)
    D[(r + 8 * half) * 128 + col0 + lr] = c[r];
}

// ---------------------------------------------------------------------------
// Kernel 3: memory-bound streaming pass. One wave per node. Each lane owns
// 4 channels: nfs as one float4, nfv[v][m] as 3 float4 (48B, 16B aligned).
// energy[n] = (1/8192) * [ <nfs, ac*R1+a*R2+c*R3+R4+is3*D*R5>
//            + is3*( <a*G1+G2, fdot> + <c*G3+G4, qdot> ) ]
// ---------------------------------------------------------------------------
__global__ void energy_kernel(const float* __restrict__ nf,
                              const float* __restrict__ c0,
                              const float* __restrict__ ci,
                              const float* __restrict__ ff,
                              const float* __restrict__ RG,   // R 16x128 then G 16x128
                              float* __restrict__ out, int N) {
  __shared__ __align__(16) float s[9 * 128];   // R1..R5, G1..G4
  const int tid = threadIdx.x;
  for (int i = tid; i < 9 * 128; i += blockDim.x) {
    const int row = i >> 7, col = i & 127;
    s[i] = (row < 5) ? RG[row * 128 + col] : RG[2048 + (row - 5) * 128 + col];
  }
  __syncthreads();

  const int wave = tid >> 5, lane = tid & 31;
  const int n = blockIdx.x * 8 + wave;
  if (n >= N) return;

  // per-node scalars (broadcast loads, L0/L2 cached)
  const float a  = ci[4*n]     - c0[4*n];
  const float q0 = ci[4*n + 1] - c0[4*n + 1];
  const float q1 = ci[4*n + 2] - c0[4*n + 2];
  const float q2 = ci[4*n + 3] - c0[4*n + 3];
  const float cc = 0.01f * ff[4*n];
  const float f0 = 0.01f * ff[4*n + 1];
  const float f1 = 0.01f * ff[4*n + 2];
  const float f2 = 0.01f * ff[4*n + 3];
  const float Dd = q0 * f0 + q1 * f1 + q2 * f2;
  const float ac = a * cc;
  const float dR5 = IS3 * Dd;

  const float4* sv = reinterpret_cast<const float4*>(s);
  const float4 R1 = sv[0*32 + lane], R2 = sv[1*32 + lane], R3 = sv[2*32 + lane];
  const float4 R4 = sv[3*32 + lane], R5 = sv[4*32 + lane];
  const float4 G1 = sv[5*32 + lane], G2 = sv[6*32 + lane];
  const float4 G3 = sv[7*32 + lane], G4 = sv[8*32 + lane];

  float4 cS, cF, cQ;
  cS.x = ac*R1.x + a*R2.x + cc*R3.x + R4.x + dR5*R5.x;
  cS.y = ac*R1.y + a*R2.y + cc*R3.y + R4.y + dR5*R5.y;
  cS.z = ac*R1.z + a*R2.z + cc*R3.z + R4.z + dR5*R5.z;
  cS.w = ac*R1.w + a*R2.w + cc*R3.w + R4.w + dR5*R5.w;
  cF.x = a*G1.x + G2.x;  cF.y = a*G1.y + G2.y;  cF.z = a*G1.z + G2.z;  cF.w = a*G1.w + G2.w;
  cQ.x = cc*G3.x + G4.x; cQ.y = cc*G3.y + G4.y; cQ.z = cc*G3.z + G4.z; cQ.w = cc*G3.w + G4.w;

  // node_feats row: 128 floats nfs, then 128x3 nfv row-major
  const float4* nrow = reinterpret_cast<const float4*>(nf + (size_t)n * 512);
  const float4 sN = nrow[lane];                 // nfs[4*lane .. 4*lane+3]
  const float4 p0 = nrow[32 + lane * 3 + 0];    // nfv channels 4*lane..4*lane+3, 12 floats
  const float4 p1 = nrow[32 + lane * 3 + 1];
  const float4 p2 = nrow[32 + lane * 3 + 2];

  const float fd0 = f0*p0.x + f1*p0.y + f2*p0.z;
  const float fd1 = f0*p0.w + f1*p1.x + f2*p1.y;
  const float fd2 = f0*p1.z + f1*p1.w + f2*p2.x;
  const float fd3 = f0*p2.y + f1*p2.z + f2*p2.w;
  const float qd0 = q0*p0.x + q1*p0.y + q2*p0.z;
  const float qd1 = q0*p0.w + q1*p1.x + q2*p1.y;
  const float qd2 = q0*p1.z + q1*p1.w + q2*p2.x;
  const float qd3 = q0*p2.y + q1*p2.z + q2*p2.w;

  float part = sN.x*cS.x + sN.y*cS.y + sN.z*cS.z + sN.w*cS.w
             + IS3 * (cF.x*fd0 + cF.y*fd1 + cF.z*fd2 + cF.w*fd3
                    + cQ.x*qd0 + cQ.y*qd1 + cQ.z*qd2 + cQ.w*qd3);

  for (int off = 16; off > 0; off >>= 1)
    part += __shfl_xor(part, off, 32);
  if (lane == 0) out[n] = FINAL_SCALE * part;
}

extern "C" void kernel_launch(void* const* d_in, const int* in_sizes, int n_in,
                              void* d_out, int out_size, void* d_ws, size_t ws_size,
                              hipStream_t stream) {
  (void)in_sizes; (void)n_in; (void)ws_size;
  const float* node_feats = (const float*)d_in[0];
  const float* c0         = (const float*)d_in[1];
  const float* ci         = (const float*)d_in[2];
  const float* ff         = (const float*)d_in[5];
  const float* wqa_s      = (const float*)d_in[8];
  const float* bqa        = (const float*)d_in[9];
  const float* wqa_v      = (const float*)d_in[10];
  const float* wva_s      = (const float*)d_in[11];
  const float* bva        = (const float*)d_in[12];
  const float* wva_v      = (const float*)d_in[13];
  const float* Wss        = (const float*)d_in[14];
  const float* Wvv        = (const float*)d_in[15];
  const float* Wsv        = (const float*)d_in[16];
  const float* Wvs        = (const float*)d_in[17];
  const float* Wd_ss      = (const float*)d_in[18];
  const float* Wd_vv      = (const float*)d_in[19];
  float* ws  = (float*)d_ws;
  float* out = (float*)d_out;
  const int N = out_size;  // 65536

  precompute_PQ<<<4, 128, 0, stream>>>(Wss, Wvv, Wsv, Wvs,
                                       wqa_s, bqa, wqa_v, wva_s, bva, wva_v, ws);
  small_gemm_wmma<<<1, 512, 0, stream>>>(ws, Wd_ss, Wd_vv, ws + 4096);
  energy_kernel<<<(N + 7) / 8, 256, 0, stream>>>(node_feats, c0, ci, ff, ws + 4096, out, N);
}